// ODEFunc_26620207301120
// MI455X (gfx1250) — compile-verified
//
#include <hip/hip_runtime.h>
#include <cstdint>

#define N_NODES 20000
#define LATENT  64
#define B_SZ    4
#define FB      256           // LATENT*B  (also UNITS*B)
#define M_STACK 5             // 2*K+1
#define KDIM    320           // LATENT*M_STACK
#define OUTF    64

typedef __attribute__((ext_vector_type(2))) float v2f;
typedef __attribute__((ext_vector_type(8))) float v8f;

// ---------------------------------------------------------------------------
// Pack input (B, N, 64) -> x0_dense (N, 256) [fb = f*4+b] and xc slice m=0
// ---------------------------------------------------------------------------
__global__ __launch_bounds__(256) void pack_x0(const float* __restrict__ inp,
                                               float* __restrict__ x0d,
                                               float* __restrict__ xc) {
    size_t idx = (size_t)blockIdx.x * 256 + threadIdx.x;   // n*256 + fb
    int n  = (int)(idx >> 8);
    int fb = (int)(idx & 255);
    int f = fb >> 2, b = fb & 3;
    float v = inp[((size_t)b * N_NODES + n) * LATENT + f];
    x0d[idx] = v;
    xc[((size_t)b * N_NODES + n) * KDIM + f * M_STACK] = v;
}

__device__ __forceinline__ int lbound(const int* __restrict__ a, int nnz, int key) {
    int lo = 0, hi = nnz;
    while (lo < hi) { int mid = (lo + hi) >> 1; if (a[mid] < key) lo = mid + 1; else hi = mid; }
    return lo;
}

// ---------------------------------------------------------------------------
// SPMM: one block per row n; 256 threads over the fb feature dim.
//   acc = sum vals[e] * xin[cols[e], fb]              (rows sorted -> bsearch)
//   if xprev:  acc = 2*acc - xprev[n, fb]             (Chebyshev step)
//   if ydense: ydense[n, fb] = acc
//   xc[(b*N+n), f*5 + mslot] = acc
// ---------------------------------------------------------------------------
__global__ __launch_bounds__(256) void spmm_step(const int* __restrict__ rows,
                                                 const int* __restrict__ cols,
                                                 const float* __restrict__ vals,
                                                 int nnz,
                                                 const float* __restrict__ xin,
                                                 const float* __restrict__ xprev,
                                                 float* __restrict__ ydense,
                                                 float* __restrict__ xc,
                                                 int mslot) {
    int n = blockIdx.x;
    __shared__ int s_lo, s_hi;
    if (threadIdx.x == 0) {
        s_lo = lbound(rows, nnz, n);
        s_hi = lbound(rows, nnz, n + 1);
    }
    __syncthreads();
    int lo = s_lo, hi = s_hi;
    int tid = threadIdx.x;
    float acc = 0.0f;
    for (int e = lo; e < hi; ++e) {
        int   c = cols[e];
        float v = vals[e];
        acc += v * xin[(size_t)c * FB + tid];          // coalesced gather (L2-resident)
    }
    if (xprev)  acc = 2.0f * acc - xprev[(size_t)n * FB + tid];
    if (ydense) ydense[(size_t)n * FB + tid] = acc;
    int f = tid >> 2, b = tid & 3;
    xc[((size_t)b * N_NODES + n) * KDIM + f * M_STACK + mslot] = acc;
}

// ---------------------------------------------------------------------------
// Fused dual GEMM (theta & hidden c), fp32 WMMA 16x16x4.
//   A  : xc (80000 x 320) row-major
//   B1 : W_lat_lat (320 x 64), bias b_lat -> theta = sigmoid(.)
//   B2 : W_lat_units (320 x 64), bias b_units -> c = tanh(.)
//   c written to c_dense (N,256 fb-layout) and xc slice m=0 (in place, after
//   __syncthreads: each block owns its 16 A-rows exclusively).
// grid = 5000 blocks of 128 threads (4 waves; wave w -> col tile w)
// ---------------------------------------------------------------------------
__global__ __launch_bounds__(128) void gemm_dual(const float* __restrict__ A,
                                                 const float* __restrict__ W1,
                                                 const float* __restrict__ b1,
                                                 const float* __restrict__ W2,
                                                 const float* __restrict__ b2,
                                                 float* __restrict__ theta,
                                                 float* __restrict__ c_dense,
                                                 float* __restrict__ xc_m0) {
    int rt   = blockIdx.x;                 // row tile: rows rt*16 .. +15
    int wave = threadIdx.x >> 5;           // col tile 0..3
    int lane = threadIdx.x & 31;
    int l16  = lane & 15;
    int kb   = (lane < 16) ? 0 : 2;        // K pair held by this lane
    int row0 = rt * 16;
    int col  = wave * 16 + l16;

    const float* arow = A + (size_t)(row0 + l16) * KDIM + kb;
    float bias1 = b1[col];
    float bias2 = b2[col];
    v8f acc1 = {bias1, bias1, bias1, bias1, bias1, bias1, bias1, bias1};
    v8f acc2 = {bias2, bias2, bias2, bias2, bias2, bias2, bias2, bias2};

    for (int k0 = 0; k0 < KDIM; k0 += 4) {
        v2f a = *reinterpret_cast<const v2f*>(arow + k0);   // A[M=l16][k0+kb, +1]
        v2f bv1, bv2;
        bv1.x = W1[(size_t)(k0 + kb)     * OUTF + col];
        bv1.y = W1[(size_t)(k0 + kb + 1) * OUTF + col];
        bv2.x = W2[(size_t)(k0 + kb)     * OUTF + col];
        bv2.y = W2[(size_t)(k0 + kb + 1) * OUTF + col];
        acc1 = __builtin_amdgcn_wmma_f32_16x16x4_f32(false, a, false, bv1,
                                                     (short)0, acc1, false, false);
        acc2 = __builtin_amdgcn_wmma_f32_16x16x4_f32(false, a, false, bv2,
                                                     (short)0, acc2, false, false);
    }

    __syncthreads();   // all waves done reading this block's 16 A-rows

    int mbase = (lane < 16) ? 0 : 8;
    for (int i = 0; i < 8; ++i) {
        int row = row0 + mbase + i;        // global row = b*N + n
        float s = 1.0f / (1.0f + __expf(-acc1[i]));
        theta[(size_t)row * OUTF + col] = s;
        float u = tanhf(acc2[i]);
        int b = row / N_NODES;
        int n = row - b * N_NODES;
        c_dense[(size_t)n * FB + col * 4 + b] = u;
        xc_m0[(size_t)row * KDIM + col * M_STACK] = u;
    }
}

// ---------------------------------------------------------------------------
// Final GEMM: out = -(theta * tanh(xc @ W_units_lat + b_lat))
// ---------------------------------------------------------------------------
__global__ __launch_bounds__(128) void gemm_final(const float* __restrict__ A,
                                                  const float* __restrict__ W,
                                                  const float* __restrict__ bias,
                                                  const float* __restrict__ theta,
                                                  float* __restrict__ out) {
    int rt   = blockIdx.x;
    int wave = threadIdx.x >> 5;
    int lane = threadIdx.x & 31;
    int l16  = lane & 15;
    int kb   = (lane < 16) ? 0 : 2;
    int row0 = rt * 16;
    int col  = wave * 16 + l16;

    const float* arow = A + (size_t)(row0 + l16) * KDIM + kb;
    float bv = bias[col];
    v8f acc = {bv, bv, bv, bv, bv, bv, bv, bv};

    for (int k0 = 0; k0 < KDIM; k0 += 4) {
        v2f a = *reinterpret_cast<const v2f*>(arow + k0);
        v2f b;
        b.x = W[(size_t)(k0 + kb)     * OUTF + col];
        b.y = W[(size_t)(k0 + kb + 1) * OUTF + col];
        acc = __builtin_amdgcn_wmma_f32_16x16x4_f32(false, a, false, b,
                                                    (short)0, acc, false, false);
    }

    int mbase = (lane < 16) ? 0 : 8;
    for (int i = 0; i < 8; ++i) {
        int row = row0 + mbase + i;
        float u = tanhf(acc[i]);
        size_t o = (size_t)row * OUTF + col;   // == b*(N*64) + n*64 + f
        out[o] = -(theta[o] * u);
    }
}

// ---------------------------------------------------------------------------
extern "C" void kernel_launch(void* const* d_in, const int* in_sizes, int n_in,
                              void* d_out, int out_size, void* d_ws, size_t ws_size,
                              hipStream_t stream) {
    const float* inp   = (const float*)d_in[0];
    // d_in[1] = t_local (unused)
    const int*   rows1 = (const int*)d_in[2];
    const int*   cols1 = (const int*)d_in[3];
    const float* vals1 = (const float*)d_in[4];
    const int*   rows2 = (const int*)d_in[5];
    const int*   cols2 = (const int*)d_in[6];
    const float* vals2 = (const float*)d_in[7];
    const float* Wll   = (const float*)d_in[8];
    const float* bl    = (const float*)d_in[9];
    const float* Wlu   = (const float*)d_in[10];
    const float* bu    = (const float*)d_in[11];
    const float* Wul   = (const float*)d_in[12];
    int nnz1 = in_sizes[2];
    int nnz2 = in_sizes[5];

    float* ws    = (float*)d_ws;
    float* xc    = ws;                                  // 25,600,000 f32
    float* x0d   = xc  + (size_t)(B_SZ * N_NODES) * KDIM; //  5,120,000 (also c_dense)
    float* x1d   = x0d + (size_t)N_NODES * FB;            //  5,120,000
    float* theta = x1d + (size_t)N_NODES * FB;            //  5,120,000

    dim3 bs256(256), bs128(128);

    // --- diffusion stack 1 (input x) ---
    pack_x0<<<N_NODES, bs256, 0, stream>>>(inp, x0d, xc);
    spmm_step<<<N_NODES, bs256, 0, stream>>>(rows1, cols1, vals1, nnz1, x0d, nullptr, x1d, xc, 1);
    spmm_step<<<N_NODES, bs256, 0, stream>>>(rows1, cols1, vals1, nnz1, x1d, x0d, nullptr, xc, 2);
    spmm_step<<<N_NODES, bs256, 0, stream>>>(rows2, cols2, vals2, nnz2, x0d, nullptr, x1d, xc, 3);
    spmm_step<<<N_NODES, bs256, 0, stream>>>(rows2, cols2, vals2, nnz2, x1d, x0d, nullptr, xc, 4);

    // --- fused theta + hidden-layer GEMMs; writes c into x0d and xc[m=0] ---
    gemm_dual<<<(B_SZ * N_NODES) / 16, bs128, 0, stream>>>(xc, Wll, bl, Wlu, bu,
                                                           theta, x0d, xc);

    // --- diffusion stack 2 (input c) ---
    spmm_step<<<N_NODES, bs256, 0, stream>>>(rows1, cols1, vals1, nnz1, x0d, nullptr, x1d, xc, 1);
    spmm_step<<<N_NODES, bs256, 0, stream>>>(rows1, cols1, vals1, nnz1, x1d, x0d, nullptr, xc, 2);
    spmm_step<<<N_NODES, bs256, 0, stream>>>(rows2, cols2, vals2, nnz2, x0d, nullptr, x1d, xc, 3);
    spmm_step<<<N_NODES, bs256, 0, stream>>>(rows2, cols2, vals2, nnz2, x1d, x0d, nullptr, xc, 4);

    // --- final GEMM fused with tanh, theta multiply, negate ---
    gemm_final<<<(B_SZ * N_NODES) / 16, bs128, 0, stream>>>(xc, Wul, bl, theta,
                                                            (float*)d_out);
}